// FNO2d_48584670052446
// MI455X (gfx1250) — compile-verified
//
#include <hip/hip_runtime.h>
#include <math.h>

// ---------------------------------------------------------------------------
// FNO2d forward for MI455X (gfx1250, wave32).
//  - conv1x1 / lifting / decoder GEMMs: V_WMMA_F32_16X16X4_F32, branch-free
//    inner loop, A-tile staged through LDS with coalesced float4 loads.
//  - spectral conv: exact mode-truncated separable DFT (only 24x12 rfft2
//    modes survive), LDS twiddle tables, integer (k*n) mod 201 phase.
//  - groupnorm: two-pass, LDS tree reduce + float atomics into d_ws stats.
// ---------------------------------------------------------------------------

#define B_        8
#define S_        192
#define HP        201                   // padded spatial extent (192 + 9)
#define WD        64
#define M1        12                    // retained modes per axis
#define NPIX_PAD  (B_ * HP * HP)        // 323208 (actual rows)
#define MPAD_LAYER 323216               // 323208 rounded up to multiple of 16
#define NPIX_CROP (B_ * S_ * S_)        // 294912 (already multiple of 16)
#define KLIFT     44                    // 43 features zero-padded to 44
#define PI_F      3.14159265358979323846f
#define TWO_PI_F  6.28318530717958647692f

typedef float v2f __attribute__((ext_vector_type(2)));
typedef float v8f __attribute__((ext_vector_type(8)));

__device__ __forceinline__ float gelu_f(float x) {
    return 0.5f * x * (1.0f + erff(x * 0.70710678118654752f));
}

// exact phase reduction: e^{sign * 2*pi*i * num / 201}
__device__ __forceinline__ float2 twiddle201(int num, float sign) {
    int p = num % HP;
    float ang = sign * TWO_PI_F * (float)p / (float)HP;
    float s, c;
    sincosf(ang, &s, &c);
    return make_float2(c, s);
}

// ---------------------------------------------------------------------------
// Feature build: [x, gridx, gridy, cos(20), sin(20), 0pad] = 44 channels.
// ---------------------------------------------------------------------------
__global__ void feat_kernel(const float* __restrict__ x, float* __restrict__ feat) {
    int idx = blockIdx.x * blockDim.x + threadIdx.x;
    if (idx >= NPIX_CROP) return;
    int r  = idx % (S_ * S_);
    int y  = r / S_;
    int w  = r % S_;
    float gx = (float)y * (1.0f / (float)(S_ - 1));
    float gy = (float)w * (1.0f / (float)(S_ - 1));
    float* f = feat + (size_t)idx * KLIFT;
    f[0] = x[idx];
    f[1] = gx;
    f[2] = gy;
    float fr = PI_F;
#pragma unroll
    for (int l = 0; l < 10; ++l) {
        float sx, cx, sy, cy;
        sincosf(gx * fr, &sx, &cx);
        sincosf(gy * fr, &sy, &cy);
        f[3  + 2 * l] = cx;
        f[4  + 2 * l] = cy;
        f[23 + 2 * l] = sx;
        f[24 + 2 * l] = sy;
        fr *= 2.0f;
    }
    f[43] = 0.f;                         // K padding column
}

// Zero-padded copy of Wp: (43,64) -> (44,64), row 43 = 0.
__global__ void wp_pad_kernel(const float* __restrict__ Wp, float* __restrict__ WpPad) {
    int i = blockIdx.x * blockDim.x + threadIdx.x;
    if (i >= KLIFT * WD) return;
    int k = i / WD;
    WpPad[i] = (k < 43) ? Wp[i] : 0.f;
}

// ---------------------------------------------------------------------------
// WMMA f32 GEMM:  C[Mpad,N] = A[Mpad,K] * W + bias.
// WNK=1: W[n*K+k] (conv 'oi' weights);  WNK=0: W[k*N+n] (lifting).
// K multiple of 4, Mpad multiple of 16, N multiple of 16. No per-element
// guards: buffers are padded at allocation; pad rows hold garbage that is
// never consumed (D rows depend only on the same A rows).
// A tile (16 x K, contiguous 64*K bytes) is staged into a wave-private LDS
// slice with coalesced float4 loads, then fragments come from ds_load_b64.
// ---------------------------------------------------------------------------
#define GEMM_WAVES 4

template <int WNK>
__global__ void gemm_wmma_kernel(const float* __restrict__ A,
                                 const float* __restrict__ W,
                                 const float* __restrict__ bias,
                                 float* __restrict__ C,
                                 int Mpad, int N, int K) {
    extern __shared__ float smAll[];     // GEMM_WAVES * 16 * K floats
    int lane = threadIdx.x & 31;
    int wave = threadIdx.x >> 5;
    int row0 = (blockIdx.x * GEMM_WAVES + wave) * 16;
    if (row0 >= Mpad) return;            // wave-uniform: EXEC stays all-1
    int n0 = blockIdx.y * 16;

    // ---- stage contiguous 16xK A tile into this wave's LDS slice ----
    float* smw = smAll + wave * 16 * K;
    {
        const float4* src = (const float4*)(A + (size_t)row0 * K);
        float4*       dst = (float4*)smw;
        int nvec = 4 * K;                // (16*K)/4 float4's
        for (int t = lane; t < nvec; t += 32) dst[t] = src[t];
    }

    int half = lane >> 4;                // 0: K = k0,k0+1   1: K = k0+2,k0+3
    int mr   = lane & 15;
    int n    = n0 + mr;

    v8f acc = {0.f, 0.f, 0.f, 0.f, 0.f, 0.f, 0.f, 0.f};

#pragma unroll 4
    for (int k0 = 0; k0 < K; k0 += 4) {
        int ka = k0 + 2 * half;
        v2f a = *(const v2f*)(smw + mr * K + ka);          // ds_load_b64
        v2f b;
        if (WNK) {
            b = *(const v2f*)(W + (size_t)n * K + ka);     // global b64 (cached)
        } else {
            b.x = W[(size_t)ka * N + n];
            b.y = W[(size_t)(ka + 1) * N + n];
        }
        acc = __builtin_amdgcn_wmma_f32_16x16x4_f32(
            false, a, false, b, (short)0, acc, false, false);
    }

    // D layout: VGPR r -> M = row0 + r + 8*half, N = n0 + (lane & 15)
#pragma unroll
    for (int r = 0; r < 8; ++r) {
        int mm = row0 + r + 8 * half;
        C[(size_t)mm * N + n] = acc[r] + bias[n];
    }
}

// ---------------------------------------------------------------------------
// Pad-scatter 192x192 -> zero-padded 201x201, and crop back.
// ---------------------------------------------------------------------------
__global__ void pad_scatter_kernel(const float* __restrict__ src, float* __restrict__ dst) {
    long idx = (long)blockIdx.x * blockDim.x + threadIdx.x;
    long total = (long)NPIX_PAD * WD;
    if (idx >= total) return;
    int  c = (int)(idx % WD);
    long p = idx / WD;
    int  b = (int)(p / (HP * HP));
    int  r = (int)(p % (HP * HP));
    int  y = r / HP, w = r % HP;
    float v = 0.f;
    if (y < S_ && w < S_)
        v = src[((size_t)(b * S_ + y) * S_ + w) * WD + c];
    dst[idx] = v;
}

__global__ void crop_kernel(const float* __restrict__ src, float* __restrict__ dst) {
    long idx = (long)blockIdx.x * blockDim.x + threadIdx.x;
    long total = (long)NPIX_CROP * WD;
    if (idx >= total) return;
    int  c = (int)(idx % WD);
    long p = idx / WD;
    int  b = (int)(p / (S_ * S_));
    int  r = (int)(p % (S_ * S_));
    int  y = r / S_, w = r % S_;
    dst[idx] = src[((size_t)(b * HP + y) * HP + w) * WD + c];
}

// ---------------------------------------------------------------------------
// Spectral conv, stage A: real DFT along W, keep ky = 0..11.
// ---------------------------------------------------------------------------
__global__ void dftW_kernel(const float* __restrict__ H, float2* __restrict__ FW) {
    int y = blockIdx.x, ky = blockIdx.y, b = blockIdx.z;
    int c = threadIdx.x;
    __shared__ float2 tw[HP];
    for (int w = c; w < HP; w += 64) tw[w] = twiddle201(ky * w, -1.0f);
    __syncthreads();
    const float* row = H + ((size_t)(b * HP + y) * HP) * WD + c;
    float re = 0.f, im = 0.f;
    for (int w = 0; w < HP; ++w) {
        float v = row[(size_t)w * WD];
        re += v * tw[w].x;
        im += v * tw[w].y;
    }
    FW[((size_t)(b * HP + y) * M1 + ky) * WD + c] = make_float2(re, im);
}

// Stage B: complex DFT along H, keep kx = 0..11 and 189..200 (24 rows).
__global__ void dftH_kernel(const float2* __restrict__ FW, float2* __restrict__ XF) {
    int kxi = blockIdx.x, ky = blockIdx.y, b = blockIdx.z;
    int c = threadIdx.x;
    int kx = (kxi < M1) ? kxi : kxi + (HP - 2 * M1);
    __shared__ float2 tw[HP];
    for (int y = c; y < HP; y += 64) tw[y] = twiddle201(kx * y, -1.0f);
    __syncthreads();
    float re = 0.f, im = 0.f;
    for (int y = 0; y < HP; ++y) {
        float2 v = FW[((size_t)(b * HP + y) * M1 + ky) * WD + c];
        float2 t = tw[y];
        re += v.x * t.x - v.y * t.y;
        im += v.x * t.y + v.y * t.x;
    }
    XF[((size_t)(b * 24 + kxi) * M1 + ky) * WD + c] = make_float2(re, im);
}

// Stage C: per-mode complex channel mix (einsum bixy,ioxy->boxy).
__global__ void modemix_kernel(const float2* __restrict__ XF,
                               const float* __restrict__ sw1,
                               const float* __restrict__ sw2,
                               float2* __restrict__ OF) {
    int mode = blockIdx.x;               // 0..287
    int kxi = mode / M1, ky = mode % M1;
    int b = blockIdx.y;
    int o = threadIdx.x;
    __shared__ float2 xin[WD];
    xin[o] = XF[((size_t)(b * 24 + kxi) * M1 + ky) * WD + o];
    __syncthreads();
    const float* wgt = (kxi < M1) ? sw1 : sw2;
    int kx = (kxi < M1) ? kxi : kxi - M1;
    float re = 0.f, im = 0.f;
    for (int i = 0; i < WD; ++i) {
        size_t off = ((((size_t)i * WD + o) * M1 + kx) * M1 + ky) * 2;
        float wr = wgt[off], wi = wgt[off + 1];
        float2 v = xin[i];
        re += v.x * wr - v.y * wi;
        im += v.x * wi + v.y * wr;
    }
    OF[((size_t)(b * 24 + kxi) * M1 + ky) * WD + o] = make_float2(re, im);
}

// Stage D: inverse complex DFT along H over the 24 retained rows.
__global__ void idftH_kernel(const float2* __restrict__ OF, float2* __restrict__ YH) {
    int y = blockIdx.x, ky = blockIdx.y, b = blockIdx.z;
    int o = threadIdx.x;
    __shared__ float2 tw[24];
    if (o < 24) {
        int kx = (o < M1) ? o : o + (HP - 2 * M1);
        tw[o] = twiddle201(kx * y, +1.0f);
    }
    __syncthreads();
    float re = 0.f, im = 0.f;
#pragma unroll
    for (int kxi = 0; kxi < 24; ++kxi) {
        float2 v = OF[((size_t)(b * 24 + kxi) * M1 + ky) * WD + o];
        float2 t = tw[kxi];
        re += v.x * t.x - v.y * t.y;
        im += v.x * t.y + v.y * t.x;
    }
    YH[((size_t)(b * HP + y) * M1 + ky) * WD + o] = make_float2(re, im);
}

// Stage E: inverse real DFT along W (Hermitian columns ky>=1 doubled),
// 1/(H*W) normalization folded into the LDS twiddle table.
__global__ void idftW_kernel(const float2* __restrict__ YH, float* __restrict__ X1) {
    int y = blockIdx.x, b = blockIdx.y;
    int o = threadIdx.x;
    __shared__ float2 tw[HP * M1];
    const float norm = 1.0f / ((float)HP * (float)HP);
    for (int t = o; t < HP * M1; t += 64) {
        int w = t / M1, ky = t % M1;
        float2 e = twiddle201(ky * w, +1.0f);
        float s = (ky == 0 ? 1.0f : 2.0f) * norm;
        tw[t] = make_float2(e.x * s, e.y * s);
    }
    __syncthreads();
    float2 yk[M1];
#pragma unroll
    for (int ky = 0; ky < M1; ++ky)
        yk[ky] = YH[((size_t)(b * HP + y) * M1 + ky) * WD + o];
    float* out = X1 + ((size_t)(b * HP + y) * HP) * WD + o;
    for (int w = 0; w < HP; ++w) {
        const float2* t = &tw[w * M1];
        float acc = 0.f;
#pragma unroll
        for (int ky = 0; ky < M1; ++ky)
            acc += yk[ky].x * t[ky].x - yk[ky].y * t[ky].y;
        out[(size_t)w * WD] = acc;
    }
}

// ---------------------------------------------------------------------------
// GroupNorm (groups = 2): zero stats -> tree-reduce sums -> normalize(+GELU).
// stats layout: [b*4 + g*2 + {sum, sumsq}]
// ---------------------------------------------------------------------------
__global__ void zero_kernel(float* p, int n) {
    int i = blockIdx.x * blockDim.x + threadIdx.x;
    if (i < n) p[i] = 0.f;
}

__global__ void gn_reduce_kernel(const float* __restrict__ X, float* __restrict__ stats,
                                 int rowsPerBatch, int C) {
    int b = blockIdx.y;
    long perBatch = (long)rowsPerBatch * C;
    const float* Xb = X + (long)b * perBatch;
    int half = C >> 1;
    float s0 = 0.f, q0 = 0.f, s1 = 0.f, q1 = 0.f;
    for (long e = (long)blockIdx.x * blockDim.x + threadIdx.x; e < perBatch;
         e += (long)gridDim.x * blockDim.x) {
        float v = Xb[e];
        int c = (int)(e % C);
        if (c < half) { s0 += v; q0 += v * v; } else { s1 += v; q1 += v * v; }
    }
    __shared__ float red[256][4];
    red[threadIdx.x][0] = s0; red[threadIdx.x][1] = q0;
    red[threadIdx.x][2] = s1; red[threadIdx.x][3] = q1;
    __syncthreads();
    for (int s = 128; s > 0; s >>= 1) {
        if ((int)threadIdx.x < s) {
#pragma unroll
            for (int j = 0; j < 4; ++j)
                red[threadIdx.x][j] += red[threadIdx.x + s][j];
        }
        __syncthreads();
    }
    if (threadIdx.x == 0) {
        atomicAdd(&stats[b * 4 + 0], red[0][0]);
        atomicAdd(&stats[b * 4 + 1], red[0][1]);
        atomicAdd(&stats[b * 4 + 2], red[0][2]);
        atomicAdd(&stats[b * 4 + 3], red[0][3]);
    }
}

__global__ void gn_apply_kernel(float* __restrict__ X, const float* __restrict__ stats,
                                const float* __restrict__ gamma, const float* __restrict__ beta,
                                int rowsPerBatch, int C, int dogelu, long total) {
    long i = (long)blockIdx.x * blockDim.x + threadIdx.x;
    if (i >= total) return;
    int  c   = (int)(i % C);
    long row = i / C;
    int  b   = (int)(row / rowsPerBatch);
    int  g   = (c >= (C >> 1)) ? 1 : 0;
    float cnt = (float)rowsPerBatch * (float)(C >> 1);
    float mu  = stats[b * 4 + g * 2] / cnt;
    float var = stats[b * 4 + g * 2 + 1] / cnt - mu * mu;
    float v = (X[i] - mu) * rsqrtf(var + 1e-5f) * gamma[c] + beta[c];
    if (dogelu) v = gelu_f(v);
    X[i] = v;
}

// h = maybe_gelu(U + V)
__global__ void add_kernel(const float* __restrict__ U, const float* __restrict__ V,
                           float* __restrict__ H, long n, int dogelu) {
    long i = (long)blockIdx.x * blockDim.x + threadIdx.x;
    if (i >= n) return;
    float v = U[i] + V[i];
    if (dogelu) v = gelu_f(v);
    H[i] = v;
}

// Final 256 -> 1 projection (N=1 is not matrix-friendly; VALU reduction).
__global__ void final_proj_kernel(const float* __restrict__ Q, const float* __restrict__ w,
                                  const float* __restrict__ b0, float* __restrict__ out) {
    int p = blockIdx.x * blockDim.x + threadIdx.x;
    if (p >= NPIX_CROP) return;
    const float4* q = (const float4*)(Q + (size_t)p * 256);
    const float4* wv = (const float4*)w;
    float acc = b0[0];
#pragma unroll 8
    for (int k = 0; k < 64; ++k) {
        float4 qa = q[k], wa = wv[k];
        acc += qa.x * wa.x + qa.y * wa.y + qa.z * wa.z + qa.w * wa.w;
    }
    out[p] = acc;
}

// ---------------------------------------------------------------------------
// Host-side helpers / launcher
// ---------------------------------------------------------------------------
static void run_gemm(const float* A, const float* W, const float* bias, float* C,
                     int Mpad, int N, int K, int wNK, hipStream_t s) {
    int mtiles = Mpad / 16;
    dim3 grid((mtiles + GEMM_WAVES - 1) / GEMM_WAVES, N / 16);
    size_t shmem = (size_t)GEMM_WAVES * 16 * K * sizeof(float);
    if (wNK)
        gemm_wmma_kernel<1><<<grid, 32 * GEMM_WAVES, shmem, s>>>(A, W, bias, C, Mpad, N, K);
    else
        gemm_wmma_kernel<0><<<grid, 32 * GEMM_WAVES, shmem, s>>>(A, W, bias, C, Mpad, N, K);
}

static void run_groupnorm(float* X, const float* gamma, const float* beta, float* stats,
                          int rowsPerBatch, int C, int dogelu, hipStream_t s) {
    zero_kernel<<<1, 32, 0, s>>>(stats, 32);
    gn_reduce_kernel<<<dim3(128, B_), 256, 0, s>>>(X, stats, rowsPerBatch, C);
    long total = (long)B_ * rowsPerBatch * C;
    gn_apply_kernel<<<(int)((total + 255) / 256), 256, 0, s>>>(X, stats, gamma, beta,
                                                               rowsPerBatch, C, dogelu, total);
}

extern "C" void kernel_launch(void* const* d_in, const int* in_sizes, int n_in,
                              void* d_out, int out_size, void* d_ws, size_t ws_size,
                              hipStream_t stream) {
    const float* x_in = (const float*)d_in[0];
    const float* Wp   = (const float*)d_in[1];
    const float* bp   = (const float*)d_in[2];
    const float* sw1  = (const float*)d_in[3];
    const float* sw2  = (const float*)d_in[4];
    const float* m1w  = (const float*)d_in[5];
    const float* m1b  = (const float*)d_in[6];
    const float* mg   = (const float*)d_in[7];
    const float* mbt  = (const float*)d_in[8];
    const float* m2w  = (const float*)d_in[9];
    const float* m2b  = (const float*)d_in[10];
    const float* ww   = (const float*)d_in[11];
    const float* wb   = (const float*)d_in[12];
    const float* ng   = (const float*)d_in[13];
    const float* nb   = (const float*)d_in[14];
    const float* q1w  = (const float*)d_in[15];
    const float* q1b  = (const float*)d_in[16];
    const float* qg   = (const float*)d_in[17];
    const float* qbt  = (const float*)d_in[18];
    const float* q2w  = (const float*)d_in[19];
    const float* q2b  = (const float*)d_in[20];
    (void)in_sizes; (void)n_in; (void)out_size; (void)ws_size;

    // Workspace carving (256B aligned).
    char* base = (char*)d_ws;
    size_t off = 0;
    auto carve = [&](size_t bytes) -> char* {
        char* p = base + off;
        off += (bytes + 255) & ~(size_t)255;
        return p;
    };
    const size_t szAct = (size_t)MPAD_LAYER * WD * sizeof(float);        // padded rows
    float*  Hh    = (float*)carve(szAct);
    float*  P1    = (float*)carve(szAct);
    float*  P2    = (float*)carve(szAct);
    float2* FW    = (float2*)carve((size_t)B_ * HP * M1 * WD * sizeof(float2));
    float2* YH    = (float2*)carve((size_t)B_ * HP * M1 * WD * sizeof(float2));
    float2* XF    = (float2*)carve((size_t)B_ * 24 * M1 * WD * sizeof(float2));
    float2* OF    = (float2*)carve((size_t)B_ * 24 * M1 * WD * sizeof(float2));
    float*  stats = (float*)carve(256);
    float*  WpPad = (float*)carve((size_t)KLIFT * WD * sizeof(float));
    float*  Q     = (float*)carve((size_t)NPIX_CROP * 256 * sizeof(float));

    // ---- Lifting: features -> h0, then zero-pad to 201x201 ------------------
    wp_pad_kernel<<<(KLIFT * WD + 255) / 256, 256, 0, stream>>>(Wp, WpPad);
    feat_kernel<<<(NPIX_CROP + 255) / 256, 256, 0, stream>>>(x_in, P2);
    run_gemm(P2, WpPad, bp, P1, NPIX_CROP, WD, KLIFT, /*wNK=*/0, stream);
    {
        long total = (long)NPIX_PAD * WD;
        pad_scatter_kernel<<<(int)((total + 255) / 256), 256, 0, stream>>>(P1, Hh);
    }

    // ---- 4 FNO layers -------------------------------------------------------
    const size_t swLayer = (size_t)WD * WD * M1 * M1 * 2;   // 1,179,648 floats
    for (int i = 0; i < 4; ++i) {
        // spectral conv: H -> X1 (into P1)
        dftW_kernel   <<<dim3(HP, M1, B_), 64, 0, stream>>>(Hh, FW);
        dftH_kernel   <<<dim3(24, M1, B_), 64, 0, stream>>>(FW, XF);
        modemix_kernel<<<dim3(24 * M1, B_), 64, 0, stream>>>(XF, sw1 + i * swLayer,
                                                             sw2 + i * swLayer, OF);
        idftH_kernel  <<<dim3(HP, M1, B_), 64, 0, stream>>>(OF, YH);
        idftW_kernel  <<<dim3(HP, B_),     64, 0, stream>>>(YH, P1);

        // mlp_block(x1): conv1x1 -> GN+GELU -> conv1x1   (P1 -> P2 -> P1)
        run_gemm(P1, m1w + i * WD * WD, m1b + i * WD, P2, MPAD_LAYER, WD, WD, 1, stream);
        run_groupnorm(P2, mg + i * WD, mbt + i * WD, stats, HP * HP, WD, /*gelu=*/1, stream);
        run_gemm(P2, m2w + i * WD * WD, m2b + i * WD, P1, MPAD_LAYER, WD, WD, 1, stream);

        // skip path: conv1x1(h) -> GN                    (H -> P2)
        run_gemm(Hh, ww + i * WD * WD, wb + i * WD, P2, MPAD_LAYER, WD, WD, 1, stream);
        run_groupnorm(P2, ng + i * WD, nb + i * WD, stats, HP * HP, WD, /*gelu=*/0, stream);

        // h = maybe_gelu(x1 + x2)
        long total = (long)NPIX_PAD * WD;
        add_kernel<<<(int)((total + 255) / 256), 256, 0, stream>>>(P1, P2, Hh, total,
                                                                   (i != 3) ? 1 : 0);
    }

    // ---- Decoder head: crop -> 64->256 GEMM -> GN+GELU -> 256->1 -----------
    {
        long total = (long)NPIX_CROP * WD;
        crop_kernel<<<(int)((total + 255) / 256), 256, 0, stream>>>(Hh, P1);
    }
    run_gemm(P1, q1w, q1b, Q, NPIX_CROP, 256, WD, 1, stream);
    run_groupnorm(Q, qg, qbt, stats, S_ * S_, 256, /*gelu=*/1, stream);
    final_proj_kernel<<<(NPIX_CROP + 255) / 256, 256, 0, stream>>>(Q, q2w, q2b, (float*)d_out);
}